// Par_Norm_88785563943272
// MI455X (gfx1250) — compile-verified
//
#include <hip/hip_runtime.h>

// Par_Norm on MI455X (gfx1250).
// B=16384 rows, D=512 features, S=16 scenes.
// out = [S,B,D] f32 (512 MB, bandwidth-bound) ++ new_moving_mean [S,D] ++ new_moving_var [S,D]

#define B_N 16384
#define D_N 512
#define S_N 16
#define NTILE (D_N / 16)   // 32 d-tiles
#define BBLK 8             // batch-chunk blocks per d-tile
#define WPB 8              // waves per block (256 threads, wave32)
#define ROWS_PER_WAVE (B_N / (BBLK * WPB))  // 256
#define KSTEPS (ROWS_PER_WAVE / 4)          // 64 WMMA K-steps of 4 rows

typedef float v2f __attribute__((ext_vector_type(2)));
typedef float v4f __attribute__((ext_vector_type(4)));
typedef float v8f __attribute__((ext_vector_type(8)));

// ---------------------------------------------------------------------------
// Kernel A: masked scene moments via V_WMMA_F32_16X16X4_F32.
//   A (16x4)  = one-hot scene mask from sid  (M=scene, K=batch row)
//   B (4x16)  = inputs tile / inputs^2 tile  (K=batch row, N=d column)
//   C (16x16) = per-(scene, d) partial sums
// A layout (ISA): lanes 0-15 M=lane, V0=K0,V1=K1; lanes 16-31 M=lane-16, V0=K2,V1=K3.
// B layout:       V0: lanes0-15 K0 / lanes16-31 K2 ; V1: K1 / K3 ; N = lane&15.
// C layout:       VGPR v, lanes0-15 -> M=v, lanes16-31 -> M=v+8 ; N = lane&15.
// ---------------------------------------------------------------------------
__global__ __launch_bounds__(256) void scene_reduce_wmma(
    const float* __restrict__ x, const int* __restrict__ sid,
    float* __restrict__ partial)
{
    const int lane  = threadIdx.x & 31;
    const int wv    = threadIdx.x >> 5;
    const int dtile = blockIdx.x;
    const int bblk  = blockIdx.y;
    const int m     = lane & 15;
    const int half  = (lane >= 16) ? 1 : 0;
    const int d     = (dtile << 4) + m;

    int b0 = (bblk * WPB + wv) * ROWS_PER_WAVE;

    v8f accS = {};   // scene sums
    v8f accQ = {};   // scene sums of squares

    #pragma unroll 4
    for (int k = 0; k < KSTEPS; ++k) {
        const int bo = b0 + 2 * half;           // batch row for V0 component
        const int s0 = sid[bo];
        const int s1 = sid[bo + 1];
        v2f a;
        a.x = (s0 == m) ? 1.0f : 0.0f;
        a.y = (s1 == m) ? 1.0f : 0.0f;

        const float* p = x + (size_t)bo * D_N + d;
        v2f bv;
        bv.x = p[0];        // row bo
        bv.y = p[D_N];      // row bo+1
        v2f bq = bv * bv;

        accS = __builtin_amdgcn_wmma_f32_16x16x4_f32(
            false, a, false, bv, (short)0, accS, false, false);
        accQ = __builtin_amdgcn_wmma_f32_16x16x4_f32(
            false, a, false, bq, (short)0, accQ, false, false);
        b0 += 4;
    }

    // Cross-wave reduction in LDS (fixed order -> deterministic).
    __shared__ float red[WPB][2][256];
    const int sbase = half * 8;
    #pragma unroll
    for (int v = 0; v < 8; ++v) {
        red[wv][0][(sbase + v) * 16 + m] = accS[v];
        red[wv][1][(sbase + v) * 16 + m] = accQ[v];
    }
    __syncthreads();

    const int t = threadIdx.x;   // element (s*16 + n), n = local d
    float r0 = 0.0f, r1 = 0.0f;
    #pragma unroll
    for (int w = 0; w < WPB; ++w) { r0 += red[w][0][t]; r1 += red[w][1][t]; }

    float* dst = partial + ((size_t)dtile * BBLK + bblk) * 2 * 256;
    dst[t]       = r0;
    dst[256 + t] = r1;
}

// ---------------------------------------------------------------------------
// Kernel B: fold partials, batch stats, EMA outputs, fused affine P/Q.
//   out[s,b,d] = x*P[s,d] + Q[s,d]
//   P = scale_com*scale_s*rsqrt(batch_var+eps), Q = off_com+off_s - batch_mean*P
// One thread per d column (512 total).
// ---------------------------------------------------------------------------
__global__ __launch_bounds__(256) void finalize_stats(
    const float* __restrict__ partial,
    const float* __restrict__ scale_com, const float* __restrict__ offset_com,
    const float* __restrict__ scale_sc,  const float* __restrict__ offset_sc,
    const float* __restrict__ mov_mean,  const float* __restrict__ mov_var,
    float* __restrict__ P, float* __restrict__ Q, float* __restrict__ out)
{
    const int dcol  = blockIdx.x * 256 + threadIdx.x;   // 0..511
    const int dtile = dcol >> 4;
    const int n     = dcol & 15;

    float ssum[S_N], ssq[S_N];
    #pragma unroll
    for (int s = 0; s < S_N; ++s) { ssum[s] = 0.0f; ssq[s] = 0.0f; }

    for (int blk = 0; blk < BBLK; ++blk) {
        const float* p = partial + ((size_t)dtile * BBLK + blk) * 2 * 256;
        #pragma unroll
        for (int s = 0; s < S_N; ++s) {
            ssum[s] += p[s * 16 + n];
            ssq[s]  += p[256 + s * 16 + n];
        }
    }

    const float invB = 1.0f / (float)B_N;
    float bmean = 0.0f, bsq = 0.0f;
    #pragma unroll
    for (int s = 0; s < S_N; ++s) {
        ssum[s] *= invB;          // scene_mean  (tf divides by full B)
        ssq[s]  *= invB;          // scene E[x^2]
        bmean += ssum[s];         // every row belongs to exactly one scene
        bsq   += ssq[s];
    }
    const float bvar    = bsq - bmean * bmean;
    const float inv_std = rsqrtf(bvar + 1e-3f);
    const float sc = scale_com[dcol];
    const float oc = offset_com[dcol];

    const size_t OFF_MM = (size_t)S_N * B_N * D_N;
    const size_t OFF_MV = OFF_MM + (size_t)S_N * D_N;

    #pragma unroll
    for (int s = 0; s < S_N; ++s) {
        const int idx = s * D_N + dcol;
        const float smean = ssum[s];
        const float svar  = ssq[s] - smean * smean;
        out[OFF_MM + idx] = mov_mean[idx] * 0.99f + smean * (1.0f - 0.99f);
        out[OFF_MV + idx] = mov_var[idx]  * 0.99f + svar  * (1.0f - 0.99f);
        const float Pv = sc * scale_sc[idx] * inv_std;
        const float Qv = oc + offset_sc[idx] - bmean * Pv;
        P[idx] = Pv;
        Q[idx] = Qv;
    }
}

// ---------------------------------------------------------------------------
// Kernel C: the 512 MB broadcast write (bandwidth-bound part).
// P/Q staged once per block in LDS (64 KB of the 320 KB WGP pool); each thread
// loads one float4 of x and writes 16 scene outputs with non-temporal b128
// stores (write-once stream; keeps the 32 MB input L2-resident).
// ---------------------------------------------------------------------------
#define ROWS_PER_BLOCK 32
__global__ __launch_bounds__(256) void broadcast_norm(
    const float* __restrict__ x, const float* __restrict__ P,
    const float* __restrict__ Q, float* __restrict__ out)
{
    __shared__ v4f Pl[S_N][D_N / 4];   // 32 KB
    __shared__ v4f Ql[S_N][D_N / 4];   // 32 KB

    const int t  = threadIdx.x & 127;  // d4 column (128 * 4 = 512 = D)
    const int rh = threadIdx.x >> 7;   // row phase 0/1

    const v4f* P4 = (const v4f*)P;
    const v4f* Q4 = (const v4f*)Q;
    for (int i = threadIdx.x; i < S_N * (D_N / 4); i += 256) {
        Pl[i >> 7][i & 127] = P4[i];
        Ql[i >> 7][i & 127] = Q4[i];
    }
    __syncthreads();

    const v4f* x4 = (const v4f*)x;
    v4f* out4 = (v4f*)out;

    int b = blockIdx.x * ROWS_PER_BLOCK + rh;
    for (int r = 0; r < ROWS_PER_BLOCK; r += 2) {
        const v4f xv = x4[(size_t)b * (D_N / 4) + t];
        #pragma unroll
        for (int s = 0; s < S_N; ++s) {
            v4f o = xv * Pl[s][t] + Ql[s][t];
            __builtin_nontemporal_store(o, &out4[((size_t)s * B_N + b) * (D_N / 4) + t]);
        }
        b += 2;
    }
}

// ---------------------------------------------------------------------------
extern "C" void kernel_launch(void* const* d_in, const int* in_sizes, int n_in,
                              void* d_out, int out_size, void* d_ws, size_t ws_size,
                              hipStream_t stream) {
    const float* x          = (const float*)d_in[0];
    const int*   sid        = (const int*)d_in[1];
    const float* scale_com  = (const float*)d_in[2];
    const float* offset_com = (const float*)d_in[3];
    const float* scale_sc   = (const float*)d_in[4];
    const float* offset_sc  = (const float*)d_in[5];
    const float* mov_mean   = (const float*)d_in[6];
    const float* mov_var    = (const float*)d_in[7];
    float* out = (float*)d_out;

    float* ws      = (float*)d_ws;
    float* partial = ws;                       // 32*8*2*256 = 131072 floats (512 KB)
    float* P       = ws + 131072;              // 16*512 = 8192 floats
    float* Q       = ws + 131072 + 8192;       // 16*512 = 8192 floats

    scene_reduce_wmma<<<dim3(NTILE, BBLK), 256, 0, stream>>>(x, sid, partial);
    finalize_stats<<<2, 256, 0, stream>>>(partial, scale_com, offset_com,
                                          scale_sc, offset_sc, mov_mean, mov_var,
                                          P, Q, out);
    broadcast_norm<<<B_N / ROWS_PER_BLOCK, 256, 0, stream>>>(x, P, Q, out);
}